// RNN_7404523619239
// MI455X (gfx1250) — compile-verified
//
#include <hip/hip_runtime.h>

#define BATCH 64
#define TLEN  2048
#define DDIM  128
#define HDIM  128
#define HSTRIDE 132   // padded LDS row stride (floats): 4*m mod 64 banks -> conflict free

typedef float v2f __attribute__((ext_vector_type(2)));
typedef float v8f __attribute__((ext_vector_type(8)));

// Out[r, n] = sum_k A[r, k] * W[n, k] + bias[n]
// A: (R,128) row-major, W: (128,128) row-major, Out: (R,128).
// One block = 16 rows; 8 waves = 8 column tiles of 16. K swept with
// V_WMMA_F32_16X16X4_F32, 4 independent accumulators to break the C->D chain.
__global__ __launch_bounds__(256) void gemm_bias_wmma(
    const float* __restrict__ A, const float* __restrict__ W,
    const float* __restrict__ bias, float* __restrict__ Out)
{
    const int lane = threadIdx.x & 31;
    const int wv   = threadIdx.x >> 5;       // column tile 0..7
    const int row0 = blockIdx.x << 4;        // 16 rows per block
    const int cb   = wv << 4;
    const int mlo  = lane & 15;              // M (A) / N (B) within tile
    const int koff = (lane >> 4) << 1;       // lanes 16..31 hold K+2,K+3
    const int mhi  = (lane >> 4) << 3;       // C/D: high half-wave = M+8

    const float* arow = A + (size_t)(row0 + mlo) * DDIM;
    const float* wrow = W + (size_t)(cb   + mlo) * DDIM;

    v8f c0 = {}, c1 = {}, c2 = {}, c3 = {};
#pragma unroll
    for (int kk = 0; kk < 32; kk += 4) {
        v2f a0 = *(const v2f*)(arow + 4 * (kk + 0) + koff);
        v2f b0 = *(const v2f*)(wrow + 4 * (kk + 0) + koff);
        v2f a1 = *(const v2f*)(arow + 4 * (kk + 1) + koff);
        v2f b1 = *(const v2f*)(wrow + 4 * (kk + 1) + koff);
        v2f a2 = *(const v2f*)(arow + 4 * (kk + 2) + koff);
        v2f b2 = *(const v2f*)(wrow + 4 * (kk + 2) + koff);
        v2f a3 = *(const v2f*)(arow + 4 * (kk + 3) + koff);
        v2f b3 = *(const v2f*)(wrow + 4 * (kk + 3) + koff);
        c0 = __builtin_amdgcn_wmma_f32_16x16x4_f32(false, a0, false, b0, (short)0, c0, false, false);
        c1 = __builtin_amdgcn_wmma_f32_16x16x4_f32(false, a1, false, b1, (short)0, c1, false, false);
        c2 = __builtin_amdgcn_wmma_f32_16x16x4_f32(false, a2, false, b2, (short)0, c2, false, false);
        c3 = __builtin_amdgcn_wmma_f32_16x16x4_f32(false, a3, false, b3, (short)0, c3, false, false);
    }
    v8f c = (c0 + c1) + (c2 + c3);

    const float bv = bias[cb + mlo];
#pragma unroll
    for (int r = 0; r < 8; ++r) {
        int m = r + mhi;
        Out[(size_t)(row0 + m) * HDIM + cb + mlo] = c[r] + bv;
    }
}

// Sequential scan: h_t = relu(xp_t + h_{t-1} @ Wh^T + bh).
// Batch split into 4 independent 16-row workgroups. Wh tile pinned in VGPRs,
// h double-buffered in LDS (padded stride), one barrier per step, xp for t
// loaded before the WMMA chain and xp for t+1 prefetched.
__global__ __launch_bounds__(256) void rnn_scan_wmma(
    const float* __restrict__ xp,   // (B,T,H) input projections (incl. bi)
    const float* __restrict__ Wh,   // (H,H)
    const float* __restrict__ bh,   // (H)
    float* __restrict__ hidden)     // (B,T,H)
{
    __shared__ float hbuf[2 * 16 * HSTRIDE];

    const int lane = threadIdx.x & 31;
    const int wv   = threadIdx.x >> 5;   // column tile 0..7
    const int cb   = wv << 4;
    const int mlo  = lane & 15;
    const int koff = (lane >> 4) << 1;
    const int mhi  = (lane >> 4) << 3;
    const int rb   = blockIdx.x << 4;    // batch-row base for this block

    // Persistent B operand: B(k, n) = Wh[cb + n][k]  (32 x v2f = 64 VGPRs)
    v2f whr[32];
    {
        const float* wrow = Wh + (size_t)(cb + mlo) * HDIM;
#pragma unroll
        for (int kk = 0; kk < 32; ++kk)
            whr[kk] = *(const v2f*)(wrow + 4 * kk + koff);
    }
    const float bv = bh[cb + mlo];

    // h_0 = 0
    for (int i = threadIdx.x; i < 16 * HSTRIDE; i += 256) hbuf[i] = 0.f;
    __syncthreads();

    for (int t = 0; t < TLEN; ++t) {
        const float* cur = hbuf + ((t    ) & 1) * (16 * HSTRIDE);
        float*       nxt = hbuf + ((t + 1) & 1) * (16 * HSTRIDE);

        // xp for this step: independent of the WMMA chain -> issue first.
        float xv[8];
        size_t gbase[8];
#pragma unroll
        for (int r = 0; r < 8; ++r) {
            int m = r + mhi;
            gbase[r] = ((size_t)(rb + m) * TLEN + t) * HDIM + cb + mlo;
            xv[r] = xp[gbase[r]];
        }
        // Overlap HBM latency of next step's xp with this step's WMMA chain.
        if (t + 1 < TLEN) {
#pragma unroll
            for (int r = 0; r < 8; ++r)
                __builtin_prefetch(&xp[gbase[r] + HDIM], 0, 3);
        }

        v8f c0 = {}, c1 = {}, c2 = {}, c3 = {};
        const float* arow = cur + mlo * HSTRIDE;
#pragma unroll
        for (int kk = 0; kk < 32; kk += 4) {
            v2f a0 = *(const v2f*)(arow + 4 * (kk + 0) + koff);
            v2f a1 = *(const v2f*)(arow + 4 * (kk + 1) + koff);
            v2f a2 = *(const v2f*)(arow + 4 * (kk + 2) + koff);
            v2f a3 = *(const v2f*)(arow + 4 * (kk + 3) + koff);
            c0 = __builtin_amdgcn_wmma_f32_16x16x4_f32(false, a0, false, whr[kk + 0], (short)0, c0, false, false);
            c1 = __builtin_amdgcn_wmma_f32_16x16x4_f32(false, a1, false, whr[kk + 1], (short)0, c1, false, false);
            c2 = __builtin_amdgcn_wmma_f32_16x16x4_f32(false, a2, false, whr[kk + 2], (short)0, c2, false, false);
            c3 = __builtin_amdgcn_wmma_f32_16x16x4_f32(false, a3, false, whr[kk + 3], (short)0, c3, false, false);
        }
        v8f c = (c0 + c1) + (c2 + c3);

#pragma unroll
        for (int r = 0; r < 8; ++r) {
            int m = r + mhi;
            float hv = c[r] + xv[r] + bv;
            hv = hv > 0.f ? hv : 0.f;
            nxt[m * HSTRIDE + cb + mlo] = hv;   // for next step
            hidden[gbase[r]] = hv;              // output
        }
        __syncthreads();
    }
}

extern "C" void kernel_launch(void* const* d_in, const int* in_sizes, int n_in,
                              void* d_out, int out_size, void* d_ws, size_t ws_size,
                              hipStream_t stream) {
    const float* seq = (const float*)d_in[0];
    const float* Wi  = (const float*)d_in[1];
    const float* bi  = (const float*)d_in[2];
    const float* Wh  = (const float*)d_in[3];
    const float* bhp = (const float*)d_in[4];
    const float* Wo  = (const float*)d_in[5];
    const float* bo  = (const float*)d_in[6];

    float* out_hidden = (float*)d_out;
    float* out_output = out_hidden + (size_t)BATCH * TLEN * HDIM;

    const int rowTiles = (BATCH * TLEN) / 16;  // 8192

    // 1) x_proj = seq @ Wi^T + bi  -> staged in the output_states region
    gemm_bias_wmma<<<rowTiles, 256, 0, stream>>>(seq, Wi, bi, out_output);
    // 2) sequential scan -> hidden_states
    rnn_scan_wmma<<<BATCH / 16, 256, 0, stream>>>(out_output, Wh, bhp, out_hidden);
    // 3) output_states = hidden @ Wo^T + bo (overwrites the staged x_proj)
    gemm_bias_wmma<<<rowTiles, 256, 0, stream>>>(out_hidden, Wo, bo, out_output);
}